// LaplaceTransformEncoder_34351148434137
// MI455X (gfx1250) — compile-verified
//
#include <hip/hip_runtime.h>
#include <hip/hip_bf16.h>
#include <math.h>

#define B_ 16
#define T_ 4096
#define D_ 128
#define K_ 256
#define NCHUNK 64
#define CHUNKLEN 64   // T_ / NCHUNK

typedef float v2f __attribute__((ext_vector_type(2)));
typedef float v8f __attribute__((ext_vector_type(8)));

__device__ __forceinline__ float softplusf(float x) {
    return x > 20.0f ? x : log1pf(__expf(x));
}

// ---------------------------------------------------------------------------
// Kernel 0: spectral norm (one block) + per-k parameter precompute.
// prm layout: [0..K) scale = gain/sigma, [K..2K) bias = b*gain,
//             [2K..3K) alpha, [3K..4K) omega
// ---------------------------------------------------------------------------
__global__ __launch_bounds__(256)
void params_kernel(const float* __restrict__ W, const float* __restrict__ bvec,
                   const float* __restrict__ u, const float* __restrict__ sr,
                   const float* __restrict__ si, const float* __restrict__ tau,
                   const float* __restrict__ bparam, float* __restrict__ prm) {
    __shared__ float red[256];
    __shared__ float vsh[D_];
    const int tid = threadIdx.x;

    // v_raw[d] = sum_k W[k,d] * u[k]   (d handled by tid < 128)
    float vv = 0.0f;
    if (tid < D_) {
        float acc = 0.0f;
        for (int k = 0; k < K_; ++k) acc += W[k * D_ + tid] * u[k];
        vv = acc;
    }
    red[tid] = vv * vv;
    __syncthreads();
    for (int s = 128; s > 0; s >>= 1) {
        if (tid < s) red[tid] += red[tid + s];
        __syncthreads();
    }
    float vnorm = fmaxf(sqrtf(red[0]), 1e-12f);
    __syncthreads();
    if (tid < D_) vsh[tid] = vv / vnorm;
    __syncthreads();

    // Wv[k] = sum_d W[k,d] * v[d]   (k = tid, all 256)
    float wv = 0.0f;
    for (int d = 0; d < D_; ++d) wv += W[tid * D_ + d] * vsh[d];
    red[tid] = wv * wv;
    __syncthreads();
    for (int s = 128; s > 0; s >>= 1) {
        if (tid < s) red[tid] += red[tid + s];
        __syncthreads();
    }
    // sigma = u2 . Wv = ||Wv||^2 / max(||Wv||, eps)
    float n2 = red[0];
    float sigma = n2 / fmaxf(sqrtf(n2), 1e-12f);
    float inv_sigma = 1.0f / sigma;

    const float tau_p = softplusf(tau[0]) + 0.001f;
    const int k = tid;
    float alpha = (softplusf(sr[k]) + 1e-6f) * tau_p;
    float omega = fminf(fmaxf(si[k] * tau_p, -(float)M_PI), (float)M_PI);
    float gain  = softplusf(bparam[k]) + 1e-8f;
    prm[k]          = gain * inv_sigma;   // scale applied to raw x@W^T
    prm[K_ + k]     = bvec[k] * gain;     // bias
    prm[2 * K_ + k] = alpha;
    prm[3 * K_ + k] = omega;
}

// ---------------------------------------------------------------------------
// Fused kernel: per (b, chunk) block of 64 timesteps x 256 modes.
//
// Phase 1 (WMMA GEMM): drive tile (64 x 256) = x[b, t0:t0+64, :] @ W^T,
//   scaled+biased, computed with V_WMMA_F32_16X16X4_F32 into LDS.
//   64 tiles of 16x16, 8 per wave, D=128 accumulated in 32 K=4 steps.
// Phase 2 (scan): thread k runs the complex linear recurrence over the
//   64 steps reading drive from LDS (conflict-free: bank = k mod 64).
//   pass 0: from z=0, track transfer product P, store (z_end, P).
//   pass 1: from init state, write cos/sin outputs.
// grid = B*NCHUNK = 1024 blocks, 256 threads (8 waves).
// ---------------------------------------------------------------------------
__global__ __launch_bounds__(256)
void fused_scan_kernel(const float* __restrict__ x, const float* __restrict__ W,
                       const float* __restrict__ dt, const float* __restrict__ prm,
                       float* __restrict__ sum, const float* __restrict__ init,
                       float* __restrict__ out, int pass) {
    __shared__ float drive_s[CHUNKLEN * K_];   // 64 KB
    __shared__ float dt_s[CHUNKLEN];

    const int b  = blockIdx.x / NCHUNK;
    const int c  = blockIdx.x % NCHUNK;
    const int t0 = c * CHUNKLEN;

    const int tid  = threadIdx.x;
    const int lane = tid & 31;
    const int wave = tid >> 5;
    const int m    = lane & 15;   // A: row / B: column within tile
    const int half = lane >> 4;

    if (tid < CHUNKLEN) dt_s[tid] = dt[(size_t)b * T_ + t0 + tid];

    // ---- Phase 1: WMMA GEMM into LDS ----
    const size_t xbase = ((size_t)b * T_ + t0) * D_;
    #pragma unroll
    for (int tile = 0; tile < CHUNKLEN * K_ / 256 / 8; ++tile) {   // 8 tiles/wave
        const int tt    = tile * 8 + wave;       // 0..63
        const int tileM = tt >> 4;               // 0..3
        const int tileN = tt & 15;               // 0..15
        const int col0  = tileN * 16;

        const float* xrow = x + xbase + (size_t)(tileM * 16 + m) * D_;
        const float* wrow = W + (size_t)(col0 + m) * D_;

        v8f acc = {};
        #pragma unroll 4
        for (int d0 = 0; d0 < D_; d0 += 4) {
            const int d = d0 + 2 * half;  // lanes 0-15: K={d0,d0+1}; 16-31: K={d0+2,d0+3}
            v2f a; a.x = xrow[d]; a.y = xrow[d + 1];
            v2f bm; bm.x = wrow[d]; bm.y = wrow[d + 1];
            acc = __builtin_amdgcn_wmma_f32_16x16x4_f32(
                      false, a, false, bm, (short)0, acc, false, false);
        }

        const int   col = col0 + m;              // C/D: N = lane&15
        const float sc  = prm[col];
        const float bs  = prm[K_ + col];
        #pragma unroll
        for (int j = 0; j < 8; ++j) {
            const int r = tileM * 16 + j + half * 8;   // C/D: M = j + 8*(lane>=16)
            drive_s[r * K_ + col] = acc[j] * sc + bs;
        }
    }
    __syncthreads();

    // ---- Phase 2: complex linear scan over the chunk ----
    const int k = tid;
    const float a = prm[2 * K_ + k];
    const float w = prm[3 * K_ + k];
    const float inv_den = 1.0f / fmaxf(a * a + w * w, 1e-6f);

    float zr, zi, Pr = 1.0f, Pi = 0.0f;
    if (pass == 0) {
        zr = 0.0f; zi = 0.0f;
    } else {
        const size_t ib = ((size_t)(b * NCHUNK + c) * K_ + k) * 2;
        zr = init[ib]; zi = init[ib + 1];
    }

    float* optr = out + ((size_t)b * T_ + t0) * (2 * K_) + k;

    for (int i = 0; i < CHUNKLEN; ++i) {
        const float d   = drive_s[i * K_ + k];
        const float dtv = dt_s[i];
        const float rho = __expf(-a * dtv);
        float st, ct;
        __sincosf(w * dtv, &st, &ct);
        const float Ar = rho * ct;
        const float Ai = rho * st;
        const float pc = (-w + rho * (a * st + w * ct)) * inv_den;
        const float ps = ( a - rho * (a * ct - w * st)) * inv_den;
        const float nzr = Ar * zr - Ai * zi + pc * d;
        const float nzi = Ai * zr + Ar * zi + ps * d;
        zr = nzr; zi = nzi;
        if (pass == 0) {
            const float nPr = Ar * Pr - Ai * Pi;
            const float nPi = Ai * Pr + Ar * Pi;
            Pr = nPr; Pi = nPi;
        } else {
            optr[(size_t)i * 2 * K_]      = zr;   // cos response
            optr[(size_t)i * 2 * K_ + K_] = zi;   // sin response
        }
    }

    if (pass == 0) {
        const size_t sb = ((size_t)(b * NCHUNK + c) * K_ + k) * 4;
        sum[sb] = zr; sum[sb + 1] = zi; sum[sb + 2] = Pr; sum[sb + 3] = Pi;
    }
}

// ---------------------------------------------------------------------------
// Combine: serial pass over the 64 chunk summaries per (b,k) -> per-chunk
// initial states. grid = B, 256 threads.
// ---------------------------------------------------------------------------
__global__ __launch_bounds__(256)
void combine_kernel(const float* __restrict__ sum, float* __restrict__ init) {
    const int b = blockIdx.x;
    const int k = threadIdx.x;
    float ir = 0.0f, ii = 0.0f;
    for (int c = 0; c < NCHUNK; ++c) {
        const size_t ib = ((size_t)(b * NCHUNK + c) * K_ + k) * 2;
        init[ib] = ir; init[ib + 1] = ii;
        const size_t sb = ((size_t)(b * NCHUNK + c) * K_ + k) * 4;
        const float zr = sum[sb],     zi = sum[sb + 1];
        const float Pr = sum[sb + 2], Pi = sum[sb + 3];
        const float nir = Pr * ir - Pi * ii + zr;
        const float nii = Pi * ir + Pr * ii + zi;
        ir = nir; ii = nii;
    }
}

extern "C" void kernel_launch(void* const* d_in, const int* in_sizes, int n_in,
                              void* d_out, int out_size, void* d_ws, size_t ws_size,
                              hipStream_t stream) {
    (void)in_sizes; (void)n_in; (void)out_size; (void)ws_size;
    const float* x      = (const float*)d_in[0];   // (B,T,D)
    const float* dt     = (const float*)d_in[1];   // (B,T)
    const float* W      = (const float*)d_in[2];   // (K,D)
    const float* bvec   = (const float*)d_in[3];   // (K,)
    const float* u      = (const float*)d_in[4];   // (K,)
    const float* sr     = (const float*)d_in[5];   // (K,)
    const float* si     = (const float*)d_in[6];   // (K,)
    const float* tau    = (const float*)d_in[7];   // (1,)
    const float* bparam = (const float*)d_in[8];   // (1,1,K)
    float* out = (float*)d_out;                    // (B,T,2K)

    float* ws   = (float*)d_ws;
    float* prm  = ws;                                      // 4*K floats
    float* sum  = ws + 4 * K_;                             // B*NCHUNK*K*4 floats (4 MiB)
    float* init = sum + (size_t)B_ * NCHUNK * K_ * 4;      // B*NCHUNK*K*2 floats (2 MiB)

    params_kernel<<<1, 256, 0, stream>>>(W, bvec, u, sr, si, tau, bparam, prm);
    fused_scan_kernel<<<B_ * NCHUNK, 256, 0, stream>>>(x, W, dt, prm, sum, init, out, 0);
    combine_kernel<<<B_, 256, 0, stream>>>(sum, init);
    fused_scan_kernel<<<B_ * NCHUNK, 256, 0, stream>>>(x, W, dt, prm, sum, init, out, 1);
}